// DynamicGraphConstruction_89318139887967
// MI455X (gfx1250) — compile-verified
//
#include <hip/hip_runtime.h>
#include <stdint.h>

// Problem constants (from reference: x is (8,16,400,512) f32)
#define BW     128          // b*w samples
#define NNODE  400
#define HFEAT  512
#define NROWS  (BW * NNODE)          // 51200 reduction rows
#define NN     (NNODE * NNODE)       // 160000 adj entries / sample
#define NZERO  48000                 // int(160000*30/100)
#define TILES_PER_DIM 25             // 400/16
#define TILES_PER_S   (TILES_PER_DIM * TILES_PER_DIM)  // 625

typedef __attribute__((ext_vector_type(2))) float v2f;
typedef __attribute__((ext_vector_type(8))) float v8f;

// Fast sigmoid: v_exp_f32 + v_add_f32 + v_rcp_f32 (no IEEE div sequence).
__device__ __forceinline__ float sigmoidf_(float x) {
    return __builtin_amdgcn_rcpf(1.0f + __expf(-x));
}

// ---------------------------------------------------------------------------
// Kernel 1: fused copy (xf) + per-row mean/max reduction.
// One wave32 per 512-float row; float4 (b128) loads/stores; shfl-xor tree.
// ---------------------------------------------------------------------------
__global__ void k_reduce_copy(const float* __restrict__ x,
                              float* __restrict__ out1,
                              float* __restrict__ g,
                              float* __restrict__ m) {
    int wave = blockIdx.x * (blockDim.x >> 5) + (threadIdx.x >> 5);
    if (wave >= NROWS) return;
    int lane = threadIdx.x & 31;
    const float4* src = (const float4*)(x    + (size_t)wave * HFEAT);
    float4*       dst = (float4*)      (out1 + (size_t)wave * HFEAT);
    float s = 0.0f, mx = -__builtin_inff();
#pragma unroll
    for (int k = 0; k < 4; ++k) {
        float4 v = src[lane + 32 * k];
        dst[lane + 32 * k] = v;                               // copy path
        s  += v.x + v.y + v.z + v.w;
        mx  = fmaxf(mx, fmaxf(fmaxf(v.x, v.y), fmaxf(v.z, v.w)));
    }
#pragma unroll
    for (int off = 16; off > 0; off >>= 1) {
        s  += __shfl_xor(s, off, 32);
        mx  = fmaxf(mx, __shfl_xor(mx, off, 32));
    }
    if (lane == 0) {
        g[wave] = s * (1.0f / (float)HFEAT);
        m[wave] = mx;
    }
}

// ---------------------------------------------------------------------------
// Kernel 2: adj = sigmoid(g ⊗ m) via V_WMMA_F32_16X16X4_F32.
// Outer product == GEMM with K=1 (padded to K=4 with zeros).
// A layout (16x4 f32): lanes 0-15 M=0..15, VGPR0 holds K=0  -> put g there.
// B layout (4x16 f32): VGPR0 lanes 0-15 hold row K=0, N=0..15 -> put m there.
// D layout (16x16 f32, 8 VGPRs): M = v + 8*(lane>=16), N = lane&15.
// ---------------------------------------------------------------------------
__global__ void k_outer_wmma(const float* __restrict__ g,
                             const float* __restrict__ m,
                             float* __restrict__ out2) {
    int wave = blockIdx.x * (blockDim.x >> 5) + (threadIdx.x >> 5);
    int s  = wave / TILES_PER_S;
    int t  = wave - s * TILES_PER_S;
    int ti = t / TILES_PER_DIM;
    int tj = t - ti * TILES_PER_DIM;
    int lane = threadIdx.x & 31;
    int li   = lane & 15;
    bool lo  = lane < 16;

    // Clamped loads for all lanes (keeps EXEC full), zero-select via cndmask.
    float gv = g[s * NNODE + ti * 16 + li];
    float mv = m[s * NNODE + tj * 16 + li];

    v2f a; a.x = lo ? gv : 0.0f; a.y = 0.0f;   // A: g in K=0 column
    v2f b; b.x = lo ? mv : 0.0f; b.y = 0.0f;   // B: m in K=0 row
    v8f c = {};
    c = __builtin_amdgcn_wmma_f32_16x16x4_f32(false, a, false, b,
                                              (short)0, c, false, false);

    int row0 = ti * 16 + ((lane >> 4) << 3);   // +8 for upper half-wave
    int col  = tj * 16 + li;
    float* o = out2 + (size_t)s * NN + col;
#pragma unroll
    for (int v = 0; v < 8; ++v)
        o[(size_t)(row0 + v) * NNODE] = sigmoidf_(c[v]);
}

// ---------------------------------------------------------------------------
// Kernel 3: per-sample radix-select of the NZERO-th smallest value.
// sigmoid > 0 => positive-float bit patterns are order-preserving, so we
// select on uint bits. 3 levels: bits [31:20], [19:8], [7:0].
// Output: T bits (value at 0-indexed rank NZERO-1) and quota of ==T ties.
// ---------------------------------------------------------------------------
__global__ void k_select(const float* __restrict__ adj,
                         uint32_t* __restrict__ tb,
                         uint32_t* __restrict__ qt) {
    __shared__ uint32_t hist[4096];
    __shared__ uint32_t sh_prefix, sh_mask, sh_remk, sh_clt;
    int s = blockIdx.x, tid = threadIdx.x;
    const float* base = adj + (size_t)s * NN;
    if (tid == 0) { sh_prefix = 0; sh_mask = 0; sh_remk = NZERO - 1; sh_clt = 0; }

    const int shifts[3] = {20, 8, 0};
    const int nbs[3]    = {4096, 4096, 256};
#pragma unroll 1
    for (int lvl = 0; lvl < 3; ++lvl) {
        int shift = shifts[lvl], nb = nbs[lvl];
        for (int i = tid; i < nb; i += blockDim.x) hist[i] = 0;
        __syncthreads();
        uint32_t pfx = sh_prefix, msk = sh_mask;
        for (int i = tid; i < NN; i += blockDim.x) {
            if (i + 2048 < NN) __builtin_prefetch(&base[i + 2048], 0, 1);
            uint32_t u = __float_as_uint(base[i]);
            if ((u & msk) == pfx)
                atomicAdd(&hist[(u >> shift) & (nb - 1)], 1u);
        }
        __syncthreads();
        if (tid == 0) {
            uint32_t cum = 0, rem = sh_remk, d = 0;
            for (d = 0; d < (uint32_t)nb; ++d) {
                uint32_t c = hist[d];
                if (cum + c > rem) break;
                cum += c;
            }
            sh_remk   = rem - cum;
            sh_clt   += cum;                       // total strictly-below count
            sh_prefix = pfx | (d << shift);
            sh_mask   = msk | ((uint32_t)(nb - 1) << shift);
        }
        __syncthreads();
    }
    if (tid == 0) { tb[s] = sh_prefix; qt[s] = (uint32_t)NZERO - sh_clt; }
}

// ---------------------------------------------------------------------------
// Kernel 4: zero entries. bits < T always; bits == T only for the first
// `quota` ties in flat-index order (matches the reference's stable sort).
// One workgroup per sample walks indices in order; wave32 ballot + LDS scan
// assigns exact index-order ranks to ties.
// ---------------------------------------------------------------------------
__global__ void k_apply(float* __restrict__ adj,
                        const uint32_t* __restrict__ tb,
                        const uint32_t* __restrict__ qt) {
    __shared__ uint32_t wcnt[8];
    __shared__ uint32_t eqbase;
    int s = blockIdx.x, tid = threadIdx.x;
    int lane = tid & 31, wv = tid >> 5;
    uint32_t T = tb[s], Q = qt[s];
    float* base = adj + (size_t)s * NN;
    if (tid == 0) eqbase = 0;
    __syncthreads();
#pragma unroll 1
    for (int i0 = 0; i0 < NN; i0 += 256) {      // 160000 = 625 * 256 exactly
        int i = i0 + tid;
        uint32_t u = __float_as_uint(base[i]);
        bool lt = (u < T);
        bool eq = (u == T);
        uint32_t bal = (uint32_t)__ballot(eq);   // wave32: low 32 bits
        uint32_t pre = __popc(bal & ((1u << lane) - 1u));
        if (lane == 0) wcnt[wv] = __popc(bal);
        __syncthreads();
        uint32_t woff = 0;
        for (int k = 0; k < wv; ++k) woff += wcnt[k];
        uint32_t rank = eqbase + woff + pre;
        if (lt || (eq && rank < Q)) base[i] = 0.0f;
        __syncthreads();
        if (tid == 0) {
            uint32_t tot = 0;
            for (int k = 0; k < 8; ++k) tot += wcnt[k];
            eqbase += tot;
        }
        __syncthreads();
    }
}

// ---------------------------------------------------------------------------
extern "C" void kernel_launch(void* const* d_in, const int* in_sizes, int n_in,
                              void* d_out, int out_size, void* d_ws, size_t ws_size,
                              hipStream_t stream) {
    (void)in_sizes; (void)n_in; (void)out_size; (void)ws_size;
    const float* x = (const float*)d_in[0];
    float* out1 = (float*)d_out;                           // xf copy
    float* out2 = out1 + (size_t)BW * NNODE * HFEAT;       // dynamic_ftr_map

    float*    g  = (float*)d_ws;            // [51200]
    float*    m  = g + NROWS;               // [51200]
    uint32_t* tb = (uint32_t*)(m + NROWS);  // [128] threshold bits
    uint32_t* qt = tb + BW;                 // [128] tie quota

    // 51200 rows, 8 waves/block -> 6400 blocks
    k_reduce_copy<<<NROWS / 8, 256, 0, stream>>>(x, out1, g, m);
    // 128 * 625 = 80000 tiles, 8 waves/block -> 10000 blocks
    k_outer_wmma<<<(BW * TILES_PER_S) / 8, 256, 0, stream>>>(g, m, out2);
    k_select<<<BW, 256, 0, stream>>>(out2, tb, qt);
    k_apply<<<BW, 256, 0, stream>>>(out2, tb, qt);
}